// LinearAttentionBlock_68650757259443
// MI455X (gfx1250) — compile-verified
//
#include <hip/hip_runtime.h>

typedef __attribute__((ext_vector_type(2))) float v2f;
typedef __attribute__((ext_vector_type(8))) float v8f;

#define NTOK 4096   // H*W
#define CDIM 256    // channels

__device__ __forceinline__ v2f ld2(const float* p) { return *(const v2f*)p; }

// ---------------------------------------------------------------------------
// C[M x 4096] = A[M x 256] @ B[256 x 4096] (+bias), f32 WMMA 16x16x4.
// REMAP (compile-time): B row kk -> (kk/64)*192 + 64 + (kk%64), i.e. pull the
// softmaxed k-rows out of the interleaved QKV buffer. Implemented as an outer
// loop over the 4 head blocks so the inner loop is branch-free with linear
// pointer increments.
// Block = 256 threads (8 waves); each wave owns one 16x16 output tile.
// grid = (N/64, M/32, B)
// ---------------------------------------------------------------------------
template <bool REMAP, bool BIAS>
__global__ __launch_bounds__(256) void gemm_f32_wmma(
    const float* __restrict__ A, long aBatch,
    const float* __restrict__ B, long bBatch,
    float* __restrict__ C, long cBatch,
    const float* __restrict__ bias)
{
    const int tid   = threadIdx.x;
    const int wave  = tid >> 5;
    const int lane  = tid & 31;
    const int l16   = lane & 15;
    const int khalf = lane >> 4;
    const int koff  = khalf * 2;

    const int b = blockIdx.z;
    const float* Ab = A + (long)b * aBatch;
    const float* Bb = B + (long)b * bBatch;
    float*       Cb = C + (long)b * cBatch;

    const int row0 = blockIdx.y * 32 + (wave >> 2) * 16;
    const int col0 = blockIdx.x * 64 + (wave & 3) * 16;
    const int col  = col0 + l16;

    v8f acc = {};
    const float* aRow = Ab + (long)(row0 + l16) * CDIM + koff;

    if (REMAP) {
        // B rows live at (hb*192 + 64 + kk), kk in [0,64) per head block hb
#pragma unroll 1
        for (int hb = 0; hb < 4; ++hb) {
            const float* ap = aRow + hb * 64;
            const float* bp = Bb + (long)(hb * 192 + 64 + koff) * NTOK + col;
            for (int k2 = 0; k2 < 64; k2 += 4) {
                v2f a = ld2(ap + k2);          // A[row, hb*64 + k2+koff .. +1]
                v2f bf;
                bf.x = bp[0];                  // B[kk,   col]
                bf.y = bp[NTOK];               // B[kk+1, col]
                acc = __builtin_amdgcn_wmma_f32_16x16x4_f32(
                          false, a, false, bf, (short)0, acc, false, false);
                bp += 4L * NTOK;
            }
        }
    } else {
        const float* bp = Bb + (long)koff * NTOK + col;
        for (int k = 0; k < CDIM; k += 4) {
            v2f a = ld2(aRow + k);
            v2f bf;
            bf.x = bp[0];
            bf.y = bp[NTOK];
            acc = __builtin_amdgcn_wmma_f32_16x16x4_f32(
                      false, a, false, bf, (short)0, acc, false, false);
            bp += 4L * NTOK;
        }
    }

    const int rbase = row0 + khalf * 8;
#pragma unroll
    for (int i = 0; i < 8; ++i) {
        int row = rbase + i;
        float v = acc[i];
        if (BIAS) v += bias[row];
        Cb[(long)row * NTOK + col] = v;
    }
}

// ---------------------------------------------------------------------------
// In-place softmax over the spatial axis for the k-rows of the QKV buffer.
// One block (256 threads) per row; 512 rows total.
// ---------------------------------------------------------------------------
__global__ __launch_bounds__(256) void softmax_k_rows(float* __restrict__ qkv)
{
    __shared__ float red[256];
    const int tid = threadIdx.x;
    const int rid = blockIdx.x;           // 0..511
    const int b   = rid >> 8;
    const int c   = rid & 255;            // h*64 + dd
    const int h   = c >> 6;
    const int dd  = c & 63;
    float* row = qkv + (long)(b * 768 + h * 192 + 64 + dd) * NTOK;

    float vals[16];
    float m = -3.402823466e38f;
#pragma unroll
    for (int i = 0; i < 16; ++i) {
        vals[i] = row[tid + i * 256];
        m = fmaxf(m, vals[i]);
    }
    red[tid] = m;
    __syncthreads();
    for (int s = 128; s > 0; s >>= 1) {
        if (tid < s) red[tid] = fmaxf(red[tid], red[tid + s]);
        __syncthreads();
    }
    m = red[0];
    __syncthreads();

    float sum = 0.f;
#pragma unroll
    for (int i = 0; i < 16; ++i) {
        vals[i] = expf(vals[i] - m);
        sum += vals[i];
    }
    red[tid] = sum;
    __syncthreads();
    for (int s = 128; s > 0; s >>= 1) {
        if (tid < s) red[tid] += red[tid + s];
        __syncthreads();
    }
    float inv = 1.0f / red[0];
#pragma unroll
    for (int i = 0; i < 16; ++i) row[tid + i * 256] = vals[i] * inv;
}

// ---------------------------------------------------------------------------
// M[b,h] (64x64) = q(64x4096) @ v^T(4096x64) * (1/4096).
// One block per (h,b); each wave reduces a 512-wide K-chunk into a 4x4 grid
// of 16x16 WMMA tiles, then waves combine via LDS atomic adds.
// ---------------------------------------------------------------------------
__global__ __launch_bounds__(256) void qvt_gemm(const float* __restrict__ qkv,
                                                float* __restrict__ Mout)
{
    __shared__ float sM[64 * 64];
    const int tid   = threadIdx.x;
    const int wave  = tid >> 5;
    const int lane  = tid & 31;
    const int l16   = lane & 15;
    const int khalf = lane >> 4;
    const int koff  = khalf * 2;
    const int h = blockIdx.x;
    const int b = blockIdx.y;

    for (int i = tid; i < 64 * 64; i += 256) sM[i] = 0.f;
    __syncthreads();

    const float* qBase = qkv + (long)(b * 768 + h * 192) * NTOK;
    const float* vBase = qBase + (long)128 * NTOK;

    v8f acc[4][4];
    {
        v8f z = {};
#pragma unroll
        for (int r = 0; r < 4; ++r)
#pragma unroll
            for (int c = 0; c < 4; ++c) acc[r][c] = z;
    }

    const int kBeg = wave * 512;
    for (int k = kBeg; k < kBeg + 512; k += 4) {
        v2f aF[4], bF[4];
#pragma unroll
        for (int r = 0; r < 4; ++r)        // A: rows = dh, K = spatial n
            aF[r] = ld2(qBase + (long)(r * 16 + l16) * NTOK + k + koff);
#pragma unroll
        for (int c = 0; c < 4; ++c)        // B: cols = dd, K = spatial n
            bF[c] = ld2(vBase + (long)(c * 16 + l16) * NTOK + k + koff);
#pragma unroll
        for (int r = 0; r < 4; ++r)
#pragma unroll
            for (int c = 0; c < 4; ++c)
                acc[r][c] = __builtin_amdgcn_wmma_f32_16x16x4_f32(
                    false, aF[r], false, bF[c], (short)0, acc[r][c], false, false);
    }

#pragma unroll
    for (int r = 0; r < 4; ++r)
#pragma unroll
        for (int c = 0; c < 4; ++c)
#pragma unroll
            for (int i = 0; i < 8; ++i) {
                int dh = r * 16 + khalf * 8 + i;
                int dd = c * 16 + l16;
                atomicAdd(&sM[dh * 64 + dd], acc[r][c][i]);
            }
    __syncthreads();

    float* out = Mout + (long)(b * 4 + h) * 4096;
    const float scale = 1.0f / (float)NTOK;   // v_w = v / N folded in here
    for (int i = tid; i < 4096; i += 256) out[i] = sM[i] * scale;
}

// ---------------------------------------------------------------------------
// W2[b][:, h*64:(h+1)*64] = w_out[:, h*64:(h+1)*64] (256x64) @ M[b,h] (64x64)
// One block per (b,h); 8 waves x (2 row-tiles x 4 col-tiles).
// ---------------------------------------------------------------------------
__global__ __launch_bounds__(256) void w2_gemm(const float* __restrict__ wOut,
                                               const float* __restrict__ Mmat,
                                               float* __restrict__ W2)
{
    const int tid   = threadIdx.x;
    const int wave  = tid >> 5;
    const int lane  = tid & 31;
    const int l16   = lane & 15;
    const int khalf = lane >> 4;
    const int koff  = khalf * 2;
    const int bh = blockIdx.x;            // b*4 + h
    const int b  = bh >> 2;
    const int h  = bh & 3;
    const float* Mb = Mmat + (long)bh * 4096;

    v8f acc[2][4];
    {
        v8f z = {};
#pragma unroll
        for (int r = 0; r < 2; ++r)
#pragma unroll
            for (int c = 0; c < 4; ++c) acc[r][c] = z;
    }

    for (int k = 0; k < 64; k += 4) {
        v2f aF[2], bF[4];
#pragma unroll
        for (int rt = 0; rt < 2; ++rt) {
            int row = (wave * 2 + rt) * 16 + l16;
            aF[rt] = ld2(wOut + (long)row * CDIM + h * 64 + k + koff);
        }
#pragma unroll
        for (int ct = 0; ct < 4; ++ct) {
            int dd = ct * 16 + l16;
            v2f t;
            t.x = Mb[(k + koff) * 64 + dd];
            t.y = Mb[(k + koff + 1) * 64 + dd];
            bF[ct] = t;
        }
#pragma unroll
        for (int rt = 0; rt < 2; ++rt)
#pragma unroll
            for (int ct = 0; ct < 4; ++ct)
                acc[rt][ct] = __builtin_amdgcn_wmma_f32_16x16x4_f32(
                    false, aF[rt], false, bF[ct], (short)0, acc[rt][ct], false, false);
    }

    float* W2b = W2 + (long)b * (CDIM * CDIM);
#pragma unroll
    for (int rt = 0; rt < 2; ++rt)
#pragma unroll
        for (int ct = 0; ct < 4; ++ct)
#pragma unroll
            for (int i = 0; i < 8; ++i) {
                int row = (wave * 2 + rt) * 16 + khalf * 8 + i;
                int col = h * 64 + ct * 16 + l16;
                W2b[(long)row * CDIM + col] = acc[rt][ct][i];
            }
}

// ---------------------------------------------------------------------------
extern "C" void kernel_launch(void* const* d_in, const int* in_sizes, int n_in,
                              void* d_out, int out_size, void* d_ws, size_t ws_size,
                              hipStream_t stream)
{
    const float* x     = (const float*)d_in[0];   // (2,256,64,64)
    const float* w_qkv = (const float*)d_in[1];   // (768,256)
    const float* w_out = (const float*)d_in[2];   // (256,256)
    const float* b_out = (const float*)d_in[3];   // (256,)
    float* out = (float*)d_out;                   // (2,256,64,64)

    float* ws   = (float*)d_ws;
    float* qkv  = ws;                             // 2*768*4096 f32 (25.2 MB)
    float* Mmat = qkv + 2L * 768 * NTOK;          // 2*4*64*64
    float* W2   = Mmat + 2L * 4 * 64 * 64;        // 2*256*256

    // 1) QKV projection: qkv[b] = w_qkv @ x[b]
    gemm_f32_wmma<false, false><<<dim3(NTOK / 64, 768 / 32, 2), 256, 0, stream>>>(
        w_qkv, 0L, x, (long)CDIM * NTOK, qkv, 768L * NTOK, nullptr);

    // 2) softmax over spatial axis on k-rows (in place)
    softmax_k_rows<<<dim3(512), 256, 0, stream>>>(qkv);

    // 3) M[b,h] = q @ v^T / N
    qvt_gemm<<<dim3(4, 2), 256, 0, stream>>>(qkv, Mmat);

    // 4) W2[b] = w_out @ blockdiag(M[b,h])
    w2_gemm<<<dim3(8), 256, 0, stream>>>(w_out, Mmat, W2);

    // 5) out[b] = W2[b] @ k_sm[b] + b_out   (k_sm rows remapped out of qkv)
    gemm_f32_wmma<true, true><<<dim3(NTOK / 64, CDIM / 32, 2), 256, 0, stream>>>(
        W2, (long)CDIM * CDIM, qkv, 768L * NTOK, out, (long)CDIM * NTOK, b_out);
}